// CondConceptSampler_22720376995980
// MI455X (gfx1250) — compile-verified
//
#include <hip/hip_runtime.h>
#include <hip/hip_bf16.h>

// ---------------- problem constants ----------------
constexpr int kC  = 8;     // cardinality (groups)
constexpr int kBW = 64;    // bottleneck width
constexpr int kSD = 64;    // state dim
constexpr int kCD = 256;   // cond dim
constexpr int kB  = 32;
constexpr int kHW = 64 * 64;   // 4096
constexpr int kT  = 32;
constexpr float kEps = 1e-5f;

// tiling / LDS strides (chosen so the two wave32 lane-halves hit disjoint
// bank ranges, and all async-B128 LDS destinations are 16B aligned)
constexpr int HBLK    = 128;          // q columns per block
constexpr int HBLOCKS = kHW / HBLK;   // 32
constexpr int WQ_STR  = 72;           // Wq^T tile     (2*72 mod 64 == 16)
constexpr int XS_STR  = 136;          // x/q tile      (2*136 mod 64 == 16, 544B rows)
constexpr int KL_STR  = 40;           // kn tile       (2*40 mod 64 == 16, 160B rows)
constexpr int SIM_STR = 33;           // sim/attn tile (scatter b32 stores)
constexpr int WL_STR  = 33;           // words tile in kproj

typedef float v2f __attribute__((ext_vector_type(2)));
typedef float v8f __attribute__((ext_vector_type(8)));

// D = A(16x4 f32) * B(4x16 f32) + C(16x16 f32), wave32, fp32-exact WMMA.
__device__ __forceinline__ v8f wmma_f32_4(v2f a, v2f b, v8f c) {
  return __builtin_amdgcn_wmma_f32_16x16x4_f32(false, a, false, b, (short)0, c,
                                               false, false);
}

// ---------------- CDNA5 async global->LDS staging --------------------------
#if defined(__gfx1250__) && \
    __has_builtin(__builtin_amdgcn_global_load_async_to_lds_b128)
#define USE_ASYNC_COPY 1
#else
#define USE_ASYNC_COPY 0
#endif

// The builtin's parameter types (from clang diagnostics):
//   arg0: int __attribute__((vector_size(16))) __attribute__((address_space(1)))*
//   arg1: LDS destination pointer
//   arg2/arg3: imm offset / imm cpol
typedef int v4i_mem __attribute__((vector_size(16)));
typedef __attribute__((address_space(1))) v4i_mem gbl_v4i;
typedef __attribute__((address_space(3))) v4i_mem lds_v4i;

// AS3 pointers are 32-bit LDS offsets: truncate the generic pointer.
// The integer detour also sheds constness legally.
__device__ __forceinline__ lds_v4i* lds_cast(void* p) {
  return (lds_v4i*)(unsigned int)(unsigned long long)p;
}
// AS1 pointers share the generic bit pattern for global memory.
__device__ __forceinline__ gbl_v4i* gbl_cast(const void* p) {
  return (gbl_v4i*)(unsigned long long)p;
}

__device__ __forceinline__ void copy_b128(const void* gsrc, void* ldst) {
#if USE_ASYNC_COPY
  // GLOBAL_LOAD_ASYNC_TO_LDS_B128: copies 16 bytes from the global address in gsrc to the LDS address in ldst, tracked by ASYNCcnt.
  __builtin_amdgcn_global_load_async_to_lds_b128(gbl_cast(gsrc), lds_cast(ldst),
                                                 0, 0);
#else
  float4 v = *(const float4*)gsrc;
  float* d = (float*)ldst;
  d[0] = v.x; d[1] = v.y; d[2] = v.z; d[3] = v.w;
#endif
}

__device__ __forceinline__ void copy_wait() {
#if USE_ASYNC_COPY
#if __has_builtin(__builtin_amdgcn_s_wait_asynccnt)
  __builtin_amdgcn_s_wait_asynccnt(0);
#else
  asm volatile("s_wait_asynccnt 0" ::: "memory");
#endif
#endif
}

// ---------------------------------------------------------------------------
// Kernel A1: k = Wk . words per (b,g) + GroupNorm(affine)  -> kbuf [B,C,SD,T]
// ---------------------------------------------------------------------------
__global__ __launch_bounds__(256) void kproj_kernel(
    const float* __restrict__ words, const float* __restrict__ Wk,
    const float* __restrict__ g2, const float* __restrict__ b2,
    float* __restrict__ kbuf) {
  __shared__ float wl[kCD * WL_STR];   // words[b]: [256 c][32 t], padded
  __shared__ float red[512];
  const int tid = threadIdx.x;
  const int bg = blockIdx.x;
  const int g = bg % kC, b = bg / kC;

  const float* wsrc = words + (size_t)b * kCD * kT;
  for (int i = tid; i < kCD * kT; i += 256) {
    int c = i >> 5, t = i & 31;
    wl[c * WL_STR + t] = wsrc[i];
  }
  __syncthreads();

  const int s = tid >> 2;             // 64 rows, 4 threads per row
  const int tq = (tid & 3) * 8;       // 8 t's per thread
  const float* wkrow = Wk + ((size_t)g * kSD + s) * kCD;
  float acc[8] = {0.f, 0.f, 0.f, 0.f, 0.f, 0.f, 0.f, 0.f};
  for (int c = 0; c < kCD; ++c) {
    float w = wkrow[c];
#pragma unroll
    for (int j = 0; j < 8; ++j) acc[j] += w * wl[c * WL_STR + tq + j];
  }
  float ssum = 0.f, ssq = 0.f;
#pragma unroll
  for (int j = 0; j < 8; ++j) { ssum += acc[j]; ssq += acc[j] * acc[j]; }
  red[tid] = ssum; red[256 + tid] = ssq;
  __syncthreads();
  for (int off = 128; off > 0; off >>= 1) {
    if (tid < off) { red[tid] += red[tid + off]; red[256 + tid] += red[256 + tid + off]; }
    __syncthreads();
  }
  const float n = (float)(kSD * kT);
  const float mean = red[0] / n;
  const float rstd = rsqrtf(red[256] / n - mean * mean + kEps);
  const float gm = g2[g * kSD + s], bt = b2[g * kSD + s];
  float* dst = kbuf + ((size_t)bg * kSD + s) * kT + tq;
#pragma unroll
  for (int j = 0; j < 8; ++j) dst[j] = (acc[j] - mean) * rstd * gm + bt;
}

// ---------------------------------------------------------------------------
// Kernel A2: L2-normalize k across groups (axis C) in place -> kn
// ---------------------------------------------------------------------------
__global__ __launch_bounds__(256) void knorm_kernel(float* __restrict__ kbuf) {
  int idx = blockIdx.x * 256 + threadIdx.x;   // B*SD*T = 65536
  int b = idx / (kSD * kT);
  int r = idx % (kSD * kT);
  int s = r >> 5, t = r & 31;
  float v[kC]; float ss = 0.f;
#pragma unroll
  for (int g = 0; g < kC; ++g) {
    v[g] = kbuf[(((size_t)(b * kC + g)) * kSD + s) * kT + t];
    ss += v[g] * v[g];
  }
  float rn = rsqrtf(ss + 1e-12f);
#pragma unroll
  for (int g = 0; g < kC; ++g)
    kbuf[(((size_t)(b * kC + g)) * kSD + s) * kT + t] = v[g] * rn;
}

// ---------------------------------------------------------------------------
// shared tile loader for the q GEMM (pass1 + pass2).
// Wq is transposed by VALU stores (scatter); the 34KB x tile goes through the
// async global->LDS DMA path (contiguous 16B chunks, 16B-aligned LDS rows).
// ---------------------------------------------------------------------------
__device__ __forceinline__ void load_q_tiles(const float* __restrict__ x,
                                             const float* __restrict__ Wq,
                                             int b, int g, int h0, int tid,
                                             float* wqT, float* xs) {
  const float* wq = Wq + (size_t)g * kSD * kBW;
  for (int i = tid; i < kSD * kBW; i += 256) {
    int s = i >> 6, c = i & 63;
    wqT[c * WQ_STR + s] = wq[i];          // transposed: wqT[c][s]
  }
  const float* xbase = x + ((size_t)(b * (kC * kBW) + g * kBW)) * kHW + h0;
  for (int i = tid; i < kBW * (HBLK / 4); i += 256) {
    int c = i >> 5, c4 = i & 31;          // 32 x 16B chunks per 128-col row
    copy_b128(xbase + (size_t)c * kHW + c4 * 4, &xs[c * XS_STR + c4 * 4]);
  }
}

// ---------------------------------------------------------------------------
// Kernel B: pass 1 — q = Wq.x via WMMA; per-block partial (sum, sumsq)
// ---------------------------------------------------------------------------
__global__ __launch_bounds__(256) void qstats_kernel(
    const float* __restrict__ x, const float* __restrict__ Wq,
    float* __restrict__ statpart) {
  __shared__ float wqT[kBW * WQ_STR];
  __shared__ float xs[kBW * XS_STR];
  __shared__ float red[512];
  const int tid = threadIdx.x;
  const int blk = blockIdx.x;
  const int hb = blk % HBLOCKS, bg = blk / HBLOCKS;
  const int g = bg % kC, b = bg / kC;

  load_q_tiles(x, Wq, b, g, hb * HBLK, tid, wqT, xs);
  copy_wait();
  __syncthreads();

  const int lane = tid & 31, wave = tid >> 5;
  const int laneN = lane & 15;
  const int kh = (lane >> 4) << 1;        // K sub-offset per lane half
  const int h0w = wave * 16;              // each wave owns a 16-column strip

  float ssum = 0.f, ssq = 0.f;
#pragma unroll
  for (int st = 0; st < 4; ++st) {        // 4 s-tiles of 16 rows
    v8f acc = {0.f, 0.f, 0.f, 0.f, 0.f, 0.f, 0.f, 0.f};
    const int sb = st * 16 + laneN;
#pragma unroll
    for (int k = 0; k < kBW; k += 4) {
      const int kk = k + kh;
      v2f a, bb;
      a.x = wqT[kk * WQ_STR + sb];
      a.y = wqT[(kk + 1) * WQ_STR + sb];
      bb.x = xs[kk * XS_STR + h0w + laneN];
      bb.y = xs[(kk + 1) * XS_STR + h0w + laneN];
      acc = wmma_f32_4(a, bb, acc);
    }
#pragma unroll
    for (int j = 0; j < 8; ++j) { float v = acc[j]; ssum += v; ssq += v * v; }
  }
  __syncthreads();
  red[tid] = ssum; red[256 + tid] = ssq;
  __syncthreads();
  for (int off = 128; off > 0; off >>= 1) {
    if (tid < off) { red[tid] += red[tid + off]; red[256 + tid] += red[256 + tid + off]; }
    __syncthreads();
  }
  if (tid == 0) {
    statpart[blk * 2 + 0] = red[0];
    statpart[blk * 2 + 1] = red[256];
  }
}

// deterministic 32-way fold of per-block stats -> stats[bg*2 + {0,1}]
__global__ void reduce_stats_kernel(const float* __restrict__ part,
                                    float* __restrict__ stats) {
  int bg = threadIdx.x;                   // one block of 256
  float s0 = 0.f, s1 = 0.f;
  for (int hb = 0; hb < HBLOCKS; ++hb) {
    s0 += part[(bg * HBLOCKS + hb) * 2 + 0];
    s1 += part[(bg * HBLOCKS + hb) * 2 + 1];
  }
  stats[bg * 2 + 0] = s0; stats[bg * 2 + 1] = s1;
}

// ---------------------------------------------------------------------------
// Kernel C: pass 2 — recompute q, GroupNorm affine, qn scale folded into
// A-fragments, sim = qn^T.kn via WMMA, masked softmax over T, column sums.
// ---------------------------------------------------------------------------
__global__ __launch_bounds__(256) void qattn_kernel(
    const float* __restrict__ x, const float* __restrict__ Wq,
    const float* __restrict__ g1, const float* __restrict__ b1,
    const unsigned char* __restrict__ mask,     // jnp.bool_: 1 byte, 1 = padded
    const float* __restrict__ kn, const float* __restrict__ stats,
    float* __restrict__ attnpart) {
  __shared__ float wqT[kBW * WQ_STR];   // later reused as sim[128][SIM_STR]
  __shared__ float xs[kBW * XS_STR];    // later reused as q[64][XS_STR]
  __shared__ float kl[kSD * KL_STR];
  __shared__ float rq[HBLK];
  __shared__ float gm[kSD], bt[kSD], validv[kT];
  const int tid = threadIdx.x;
  const int blk = blockIdx.x;
  const int hb = blk % HBLOCKS, bg = blk / HBLOCKS;
  const int g = bg % kC, b = bg / kC;

  load_q_tiles(x, Wq, b, g, hb * HBLK, tid, wqT, xs);
  {  // kn tile: [64 s][32 t] -> kl, 16B-aligned rows, async B128 chunks
    const float* ksrc = kn + (size_t)bg * kSD * kT;
    for (int i = tid; i < kSD * (kT / 4); i += 256) {
      int s = i >> 3, c4 = i & 7;
      copy_b128(ksrc + (size_t)s * kT + c4 * 4, &kl[s * KL_STR + c4 * 4]);
    }
  }
  if (tid < kSD) { gm[tid] = g1[g * kSD + tid]; bt[tid] = b1[g * kSD + tid]; }
  if (tid < kT) validv[tid] = mask[b * kT + tid] ? 0.f : 1.f;
  copy_wait();
  __syncthreads();

  const int lane = tid & 31, wave = tid >> 5;
  const int laneN = lane & 15;
  const int kh = (lane >> 4) << 1;
  const int h0w = wave * 16;

  // --- q GEMM (keep all 4 s-tiles live so xs can be overwritten after) ---
  v8f qacc[4];
#pragma unroll
  for (int st = 0; st < 4; ++st) {
    v8f acc = {0.f, 0.f, 0.f, 0.f, 0.f, 0.f, 0.f, 0.f};
    const int sb = st * 16 + laneN;
#pragma unroll
    for (int k = 0; k < kBW; k += 4) {
      const int kk = k + kh;
      v2f a, bb;
      a.x = wqT[kk * WQ_STR + sb];
      a.y = wqT[(kk + 1) * WQ_STR + sb];
      bb.x = xs[kk * XS_STR + h0w + laneN];
      bb.y = xs[(kk + 1) * XS_STR + h0w + laneN];
      acc = wmma_f32_4(a, bb, acc);
    }
    qacc[st] = acc;
  }

  // --- GroupNorm affine, stash q tile in LDS (aliases xs: column-disjoint) ---
  const float nq = (float)kSD * (float)kHW;
  const float mean = stats[bg * 2 + 0] / nq;
  const float rstd = rsqrtf(stats[bg * 2 + 1] / nq - mean * mean + kEps);
#pragma unroll
  for (int st = 0; st < 4; ++st) {
#pragma unroll
    for (int j = 0; j < 8; ++j) {
      int s = st * 16 + j + ((lane >> 4) << 3);       // D layout rows
      float v = (qacc[st][j] - mean) * rstd * gm[s] + bt[s];
      xs[s * XS_STR + h0w + laneN] = v;
    }
  }
  __syncthreads();

  // per-column rsqrt(sum_s q^2)
  if (tid < HBLK) {
    float ss = 0.f;
    for (int s = 0; s < kSD; ++s) { float v = xs[s * XS_STR + tid]; ss += v * v; }
    rq[tid] = rsqrtf(ss + 1e-12f);
  }
  __syncthreads();

  // --- sim GEMM: A = qn^T (scale folded into fragments), B = kn ---
  const float rqv = rq[h0w + laneN];
  v8f sim0 = {0.f, 0.f, 0.f, 0.f, 0.f, 0.f, 0.f, 0.f};
  v8f sim1 = {0.f, 0.f, 0.f, 0.f, 0.f, 0.f, 0.f, 0.f};
#pragma unroll
  for (int k = 0; k < kSD; k += 4) {
    const int kk = k + kh;
    v2f a, bb0, bb1;
    a.x = xs[kk * XS_STR + h0w + laneN] * rqv;
    a.y = xs[(kk + 1) * XS_STR + h0w + laneN] * rqv;
    bb0.x = kl[kk * KL_STR + laneN];
    bb0.y = kl[(kk + 1) * KL_STR + laneN];
    bb1.x = kl[kk * KL_STR + 16 + laneN];
    bb1.y = kl[(kk + 1) * KL_STR + 16 + laneN];
    sim0 = wmma_f32_4(a, bb0, sim0);
    sim1 = wmma_f32_4(a, bb1, sim1);
  }
  // scatter sim tiles into LDS (reuse wqT region: 128*33 <= 64*72)
  float* simbuf = wqT;
  const int hrow = h0w + ((lane >> 4) << 3);
#pragma unroll
  for (int j = 0; j < 8; ++j) {
    simbuf[(hrow + j) * SIM_STR + laneN] = sim0[j];
    simbuf[(hrow + j) * SIM_STR + 16 + laneN] = sim1[j];
  }
  __syncthreads();

  // masked softmax per pixel row; write attn back into simbuf
  if (tid < HBLK) {
    const int h = tid;
    float m = -3.0e38f;
#pragma unroll
    for (int t = 0; t < kT; ++t)
      if (validv[t] > 0.5f) m = fmaxf(m, simbuf[h * SIM_STR + t]);
    float p[kT]; float s = 0.f;
#pragma unroll
    for (int t = 0; t < kT; ++t) {
      float v = (validv[t] > 0.5f) ? __expf(simbuf[h * SIM_STR + t] - m) : 0.f;
      p[t] = v; s += v;
    }
    const float inv = 1.f / s;
#pragma unroll
    for (int t = 0; t < kT; ++t) simbuf[h * SIM_STR + t] = p[t] * inv;
  }
  __syncthreads();

  // deterministic column sum over the 128 pixels of this block
  if (tid < kT) {
    float s = 0.f;
    for (int h = 0; h < HBLK; ++h) s += simbuf[h * SIM_STR + tid];
    attnpart[(size_t)blk * kT + tid] = s;
  }
}

// deterministic 32-way fold: attnacc[bg*T + t] = sum_hb attnpart
__global__ __launch_bounds__(256) void reduce_attn_kernel(
    const float* __restrict__ part, float* __restrict__ attnacc) {
  int idx = blockIdx.x * 256 + threadIdx.x;   // 256 bg * 32 t = 8192
  int bg = idx >> 5, t = idx & 31;
  float s = 0.f;
  for (int hb = 0; hb < HBLOCKS; ++hb)
    s += part[((size_t)bg * HBLOCKS + hb) * kT + t];
  attnacc[bg * kT + t] = s;
}

// ctx[b,g,s] = sum_t kn * attnsum / HW
__global__ __launch_bounds__(256) void ctx_kernel(
    const float* __restrict__ kbuf, const float* __restrict__ attnacc,
    float* __restrict__ out) {
  int idx = blockIdx.x * 256 + threadIdx.x;   // B*C*SD = 16384
  int bg = idx >> 6, s = idx & 63;
  const float* kr = kbuf + ((size_t)bg * kSD + s) * kT;
  const float* ar = attnacc + (size_t)bg * kT;
  float acc = 0.f;
#pragma unroll
  for (int t = 0; t < kT; ++t) acc += kr[t] * ar[t];
  out[idx] = acc * (1.0f / (float)kHW);
}

// ---------------------------------------------------------------------------
extern "C" void kernel_launch(void* const* d_in, const int* in_sizes, int n_in,
                              void* d_out, int out_size, void* d_ws, size_t ws_size,
                              hipStream_t stream) {
  (void)in_sizes; (void)n_in; (void)out_size; (void)ws_size;
  const float* x     = (const float*)d_in[0];
  const float* words = (const float*)d_in[1];
  const unsigned char* mask = (const unsigned char*)d_in[2];
  const float* Wq = (const float*)d_in[3];
  const float* Wk = (const float*)d_in[4];
  const float* g1 = (const float*)d_in[5];
  const float* b1 = (const float*)d_in[6];
  const float* g2 = (const float*)d_in[7];
  const float* b2 = (const float*)d_in[8];
  float* out = (float*)d_out;

  // workspace layout (floats)
  float* kbuf     = (float*)d_ws;                       // B*C*SD*T = 524288
  float* statpart = kbuf + (size_t)kB * kC * kSD * kT;  // 8192 blocks * 2
  float* stats    = statpart + (size_t)kB * kC * HBLOCKS * 2;   // 512
  float* attnpart = stats + (size_t)kB * kC * 2;        // 8192 blocks * 32
  float* attnacc  = attnpart + (size_t)kB * kC * HBLOCKS * kT;  // 8192

  kproj_kernel<<<kB * kC, 256, 0, stream>>>(words, Wk, g2, b2, kbuf);
  knorm_kernel<<<(kB * kSD * kT) / 256, 256, 0, stream>>>(kbuf);
  qstats_kernel<<<kB * kC * HBLOCKS, 256, 0, stream>>>(x, Wq, statpart);
  reduce_stats_kernel<<<1, kB * kC, 0, stream>>>(statpart, stats);
  qattn_kernel<<<kB * kC * HBLOCKS, 256, 0, stream>>>(x, Wq, g1, b1, mask,
                                                      kbuf, stats, attnpart);
  reduce_attn_kernel<<<(kB * kC * kT) / 256, 256, 0, stream>>>(attnpart, attnacc);
  ctx_kernel<<<(kB * kC * kSD) / 256, 256, 0, stream>>>(kbuf, attnacc, out);
}